// SAGEConv_26216480375294
// MI455X (gfx1250) — compile-verified
//
#include <hip/hip_runtime.h>

#define D 64  // D_IN == D_OUT == 64

typedef __attribute__((ext_vector_type(2))) float v2f;
typedef __attribute__((ext_vector_type(8))) float v8f;

// ---------------------------------------------------------------------------
// Kernel 1: zero the workspace (neigh_sum [N*64] + deg [N]) every call.
// ---------------------------------------------------------------------------
__global__ void sage_zero_ws(float* __restrict__ p, long long n) {
    long long i = (long long)blockIdx.x * blockDim.x + threadIdx.x;
    if (i < n) p[i] = 0.0f;
}

// ---------------------------------------------------------------------------
// Kernel 2: edge scatter. One wave32 per edge: lane handles 2 features.
//   neigh[dst] += feat[src];  deg[dst] += 1
// feat (25.6MB) and neigh (25.6MB) are L2-resident on MI455X (192MB L2),
// so the f32 atomics resolve in L2 atomic units.
// ---------------------------------------------------------------------------
__global__ void sage_scatter(const float* __restrict__ feat,
                             const int*   __restrict__ src,
                             const int*   __restrict__ dst,
                             float* __restrict__ neigh,
                             float* __restrict__ deg,
                             int n_edges) {
    long long gid = (long long)blockIdx.x * blockDim.x + threadIdx.x;
    int e    = (int)(gid >> 5);
    int lane = (int)(gid & 31);
    if (e >= n_edges) return;

    int s = src[e];
    int d = dst[e];

    v2f f = *(const v2f*)(feat + (long long)s * D + lane * 2);
    float* base = neigh + (long long)d * D + lane * 2;
    atomicAdd(base + 0, f.x);
    atomicAdd(base + 1, f.y);
    if (lane == 0) atomicAdd(deg + d, 1.0f);
}

// ---------------------------------------------------------------------------
// Kernel 3: fused mean + dual GEMM with V_WMMA_F32_16X16X4_F32 (fp32-exact).
//   out = feat @ W_self + (neigh/deg) @ W_neigh + (b_self + b_neigh)
// One wave per 16-row strip; 4 column tiles of 16; K=64 in steps of 4.
// Weights staged in LDS in K-pair-interleaved layout:
//   sW2[(k>>1)*64 + col] = { W[k][col], W[k+1][col] }   (k even)
// so every B fragment is one aligned ds_load_b64 into an even VGPR pair
// (no repacking movs in front of the WMMAs).
// Assumes n_nodes % 16 == 0 (true here: 100000).
// ---------------------------------------------------------------------------
__global__ __launch_bounds__(256)
void sage_gemm(const float* __restrict__ feat,
               const float* __restrict__ Wself,
               const float* __restrict__ bself,
               const float* __restrict__ Wneigh,
               const float* __restrict__ bneigh,
               const float* __restrict__ neigh,
               const float* __restrict__ deg,
               float* __restrict__ out,
               int n_nodes) {
    __shared__ v2f sWs2[(D / 2) * D];   // 16 KB
    __shared__ v2f sWn2[(D / 2) * D];   // 16 KB

    // Cooperative load + K-pair interleave of both 64x64 weight matrices.
    // i indexes a float4 of row-major W: row k = i>>4, cols c..c+3 = (i&15)*4.
    for (int i = threadIdx.x; i < (D * D) / 4; i += 256) {
        float4 ws = ((const float4*)Wself)[i];
        float4 wn = ((const float4*)Wneigh)[i];
        int k = i >> 4;
        int c = (i & 15) * 4;
        int p = k >> 1;
        int o = k & 1;
        float* dsp = (float*)sWs2 + p * (D * 2) + 2 * c + o;
        float* dnp = (float*)sWn2 + p * (D * 2) + 2 * c + o;
        dsp[0] = ws.x; dsp[2] = ws.y; dsp[4] = ws.z; dsp[6] = ws.w;
        dnp[0] = wn.x; dnp[2] = wn.y; dnp[4] = wn.z; dnp[6] = wn.w;
    }
    __syncthreads();

    int lane = threadIdx.x & 31;
    int wave = threadIdx.x >> 5;
    int tile = blockIdx.x * 8 + wave;
    long long row_base = (long long)tile * 16;
    if (row_base >= n_nodes) return;   // wave-uniform: EXEC stays all-ones

    // A-fragment addressing (16x4 f32): lanes 0-15 -> K{0,1}, lanes 16-31 -> K{2,3}
    int m    = lane & 15;        // row within 16-row tile
    int half = lane >> 4;        // 0 or 1
    int koff = half << 1;        // 0 or 2
    int n0   = lane & 15;        // column within a 16-col tile (B/C layout)

    const float* arow_s = feat  + (row_base + m) * D;
    const float* arow_n = neigh + (row_base + m) * D;
    float invd = 1.0f / fmaxf(deg[row_base + m], 1.0f);   // mean folds into A

    v8f acc0 = {}, acc1 = {}, acc2 = {}, acc3 = {};

    for (int k0 = 0; k0 < D; k0 += 4) {
        int kb = k0 + koff;      // even
        int p  = kb >> 1;        // K-pair index into interleaved LDS image
        v2f a_s = *(const v2f*)(arow_s + kb);
        v2f a_r = *(const v2f*)(arow_n + kb);
        v2f a_n;
        a_n.x = a_r.x * invd;
        a_n.y = a_r.y * invd;

#define SAGE_STEP(ACC, T)                                                        \
        {                                                                        \
            int col = (T) * 16 + n0;                                             \
            v2f bs = sWs2[p * D + col];                                          \
            v2f bn = sWn2[p * D + col];                                          \
            ACC = __builtin_amdgcn_wmma_f32_16x16x4_f32(                         \
                      false, a_s, false, bs, (short)0, ACC, false, false);       \
            ACC = __builtin_amdgcn_wmma_f32_16x16x4_f32(                         \
                      false, a_n, false, bn, (short)0, ACC, false, false);       \
        }
        SAGE_STEP(acc0, 0)
        SAGE_STEP(acc1, 1)
        SAGE_STEP(acc2, 2)
        SAGE_STEP(acc3, 3)
#undef SAGE_STEP
    }

    // C/D layout: VGPR r -> row r (lanes 0-15) or row 8+r (lanes 16-31), col = n0
    long long mrow0 = row_base + half * 8;
#define SAGE_STORE(ACC, T)                                                       \
    {                                                                            \
        int col = (T) * 16 + n0;                                                 \
        float bias = bself[col] + bneigh[col];                                   \
        _Pragma("unroll")                                                        \
        for (int r = 0; r < 8; ++r)                                              \
            out[(mrow0 + r) * D + col] = ACC[r] + bias;                          \
    }
    SAGE_STORE(acc0, 0)
    SAGE_STORE(acc1, 1)
    SAGE_STORE(acc2, 2)
    SAGE_STORE(acc3, 3)
#undef SAGE_STORE
}

// ---------------------------------------------------------------------------
// Launch: inputs are {feat, W_self, b_self, W_neigh, b_neigh, src, dst}.
// Workspace layout: [neigh_sum: N*64 f32][deg: N f32]  (~26 MB).
// ---------------------------------------------------------------------------
extern "C" void kernel_launch(void* const* d_in, const int* in_sizes, int n_in,
                              void* d_out, int out_size, void* d_ws, size_t ws_size,
                              hipStream_t stream) {
    const float* feat   = (const float*)d_in[0];
    const float* Wself  = (const float*)d_in[1];
    const float* bself  = (const float*)d_in[2];
    const float* Wneigh = (const float*)d_in[3];
    const float* bneigh = (const float*)d_in[4];
    const int*   src    = (const int*)d_in[5];
    const int*   dst    = (const int*)d_in[6];

    int n_nodes = in_sizes[0] / D;
    int n_edges = in_sizes[5];
    float* out = (float*)d_out;

    float* neigh = (float*)d_ws;
    float* deg   = neigh + (long long)n_nodes * D;

    // 1) zero workspace (neigh + deg)
    long long nzero = (long long)n_nodes * (D + 1);
    sage_zero_ws<<<(int)((nzero + 255) / 256), 256, 0, stream>>>(neigh, nzero);

    // 2) edge scatter: 32 lanes per edge
    long long sthreads = (long long)n_edges * 32;
    sage_scatter<<<(int)((sthreads + 255) / 256), 256, 0, stream>>>(
        feat, src, dst, neigh, deg, n_edges);

    // 3) fused mean + dual GEMM (8 waves/block, 16 rows per wave)
    int tiles  = (n_nodes + 15) / 16;
    int blocks = (tiles + 7) / 8;
    sage_gemm<<<blocks, 256, 0, stream>>>(
        feat, Wself, bself, Wneigh, bneigh, neigh, deg, out, n_nodes);
}